// Model_64484638982400
// MI455X (gfx1250) — compile-verified
//
#include <hip/hip_runtime.h>

// Problem constants (match reference)
#define NNODES 50000
#define NEDGES 800000
#define INC    64
#define HIDC   64
#define OUTC2  32

typedef float v2f __attribute__((ext_vector_type(2)));
typedef float v4f __attribute__((ext_vector_type(4)));
typedef float v8f __attribute__((ext_vector_type(8)));

// ---------------------------------------------------------------------------
// Zero a float buffer with 128-bit stores (agg scratch must be re-zeroed every
// call: we accumulate atomically into it). n must be a multiple of 4 (it is:
// 50000*64).
// ---------------------------------------------------------------------------
__global__ void zero_f32_v4(float* __restrict__ p, int n4) {
    int i = blockIdx.x * blockDim.x + threadIdx.x;
    if (i < n4) ((v4f*)p)[i] = (v4f){0.f, 0.f, 0.f, 0.f};
}

// ---------------------------------------------------------------------------
// Edge scatter: agg[dst[e]][c] += feat[src[e]][c]
// One wave32 per edge; each lane owns 2 consecutive channels (float2 load +
// two native global_atomic_add_f32, no-return form -> STOREcnt, no CAS loop).
// feat table (12.8 MB) and agg (12.8 MB) are both L2-resident on a 192 MB L2,
// so the random gather and the atomic RMWs are absorbed by L2 atomic units.
// ---------------------------------------------------------------------------
__global__ void scatter_sum(const float* __restrict__ feat,
                            const int*   __restrict__ src,
                            const int*   __restrict__ dst,
                            float*       __restrict__ agg) {
    long long tid = (long long)blockIdx.x * blockDim.x + threadIdx.x;
    int e  = (int)(tid >> 5);        // wave-uniform edge id
    int c0 = (int)(tid & 31) * 2;    // lane's first channel
    if (e >= NEDGES) return;
    int s = src[e];
    int d = dst[e];
    v2f v = *(const v2f*)(feat + (long long)s * 64 + c0);
    float* p = agg + (long long)d * 64 + c0;
    // CDNA5 VGLOBAL atomic, no pre-op return (tracked by STOREcnt).
    asm volatile("global_atomic_add_f32 %0, %1, off" :: "v"(p),     "v"(v.x) : "memory");
    asm volatile("global_atomic_add_f32 %0, %1, off" :: "v"(p + 1), "v"(v.y) : "memory");
}

// ---------------------------------------------------------------------------
// Fused dual-GEMM + bias (+ optional ReLU):
//   Out = act( A1 @ W1 + A2 @ W2 + b )       A*: [nRows x 64], W*: [64 x OUTC]
// One wave computes one 16x16 output tile with V_WMMA_F32_16X16X4_F32,
// chaining the accumulator across both products (K=64 -> 16 steps x 2 mats).
// Weight fragments for the wave's column tile are preloaded once into VGPRs
// (2 mats x 16 ksteps x v2f = 64 VGPRs) and reused across all row tiles.
//
// Fragment layouts per CDNA5 ISA 7.12.2 (wave32):
//   A (16x4 f32):  lane<16: M=lane,   holds K=k,k+1 ; lane>=16: M=lane-16, K=k+2,k+3
//   B (4x16 f32):  lane<16: N=lane,   holds K=k,k+1 ; lane>=16: N=lane-16, K=k+2,k+3
//   C/D (16x16):   VGPR i, lane<16 -> (row i,    col lane)
//                          lane>=16 -> (row i+8, col lane-16)
// ---------------------------------------------------------------------------
template <int OUTC, bool RELU>
__global__ void gcn_gemm_wmma(const float* __restrict__ A1,
                              const float* __restrict__ W1,
                              const float* __restrict__ A2,
                              const float* __restrict__ W2,
                              const float* __restrict__ bias,
                              float*       __restrict__ Out,
                              int nRows) {
    constexpr int NCT    = OUTC / 16;   // column tiles
    constexpr int KSTEPS = INC / 4;     // 16 wmma K-steps

    const int lane = threadIdx.x & 31;
    const int half = lane >> 4;         // 0: K pair {0,1}, 1: K pair {2,3}
    const int lid  = lane & 15;

    const int wid    = (blockIdx.x * blockDim.x + threadIdx.x) >> 5;
    const int nWaves = (gridDim.x * blockDim.x) >> 5;

    const int ct       = wid % NCT;
    const int rt0      = wid / NCT;
    const int rtStride = nWaves / NCT;  // launch keeps nWaves % NCT == 0

    const int c  = ct * 16 + lid;       // this lane's output column
    const float bb = bias[c];

    // Preload B fragments for both weight matrices (reused for every row tile)
    v2f bw1[KSTEPS], bw2[KSTEPS];
#pragma unroll
    for (int kk = 0; kk < KSTEPS; ++kk) {
        const int kr = 4 * kk + 2 * half;
        bw1[kk].x = W1[(kr    ) * OUTC + c];
        bw1[kk].y = W1[(kr + 1) * OUTC + c];
        bw2[kk].x = W2[(kr    ) * OUTC + c];
        bw2[kk].y = W2[(kr + 1) * OUTC + c];
    }

    const int nRowTiles = (nRows + 15) >> 4;
    for (int rt = rt0; rt < nRowTiles; rt += rtStride) {
        // Lane's A row; clamp (branch-free) so EXEC stays all-1s for WMMA.
        const int row  = rt * 16 + lid;
        const int rowc = min(row, nRows - 1);
        const float* a1p = A1 + (long long)rowc * INC + 2 * half;
        const float* a2p = A2 + (long long)rowc * INC + 2 * half;

        v8f acc;
#pragma unroll
        for (int i = 0; i < 8; ++i) acc[i] = bb;   // bias broadcast

#pragma unroll
        for (int kk = 0; kk < KSTEPS; ++kk) {
            v2f a1 = *(const v2f*)(a1p + 4 * kk);
            v2f a2 = *(const v2f*)(a2p + 4 * kk);
            acc = __builtin_amdgcn_wmma_f32_16x16x4_f32(
                false, a1, false, bw1[kk], (short)0, acc, false, false);
            acc = __builtin_amdgcn_wmma_f32_16x16x4_f32(
                false, a2, false, bw2[kk], (short)0, acc, false, false);
        }

        // Writeback (divergence after the WMMA region is fine)
#pragma unroll
        for (int i = 0; i < 8; ++i) {
            const int r = rt * 16 + i + 8 * half;
            if (r < nRows) {
                float v = acc[i];
                if (RELU) v = fmaxf(v, 0.0f);
                Out[(long long)r * OUTC + c] = v;
            }
        }
    }
}

// ---------------------------------------------------------------------------
// Launch: zero agg -> scatter(x) -> layer1 WMMA (relu) ->
//         zero agg -> scatter(h) -> layer2 WMMA
// ws layout: [agg: 50000*64 f32][h: 50000*64 f32]  (25.6 MB)
// ---------------------------------------------------------------------------
extern "C" void kernel_launch(void* const* d_in, const int* in_sizes, int n_in,
                              void* d_out, int out_size, void* d_ws, size_t ws_size,
                              hipStream_t stream) {
    const float* x      = (const float*)d_in[0];
    const int*   edges  = (const int*)d_in[1];          // [2, NEDGES]
    const float* Wrel1  = (const float*)d_in[2];
    const float* Wroot1 = (const float*)d_in[3];
    const float* b1     = (const float*)d_in[4];
    const float* Wrel2  = (const float*)d_in[5];
    const float* Wroot2 = (const float*)d_in[6];
    const float* b2     = (const float*)d_in[7];
    float*       out    = (float*)d_out;

    const int* src = edges;            // edge_index[0]
    const int* dst = edges + NEDGES;   // edge_index[1]

    float* agg = (float*)d_ws;
    float* h   = agg + (size_t)NNODES * HIDC;

    const int aggN4      = (NNODES * HIDC) / 4;              // 800k float4
    const int zeroBlocks = (aggN4 + 255) / 256;
    const long long scatThreads = (long long)NEDGES * 32;    // 25.6M
    const int scatBlocks = (int)((scatThreads + 255) / 256);

    // Layer 1
    zero_f32_v4<<<zeroBlocks, 256, 0, stream>>>(agg, aggN4);
    scatter_sum<<<scatBlocks, 256, 0, stream>>>(x, src, dst, agg);
    // 256 thr = 8 waves/block (multiple of NCT=4); 512 blocks = 4096 waves
    gcn_gemm_wmma<HIDC, true><<<512, 256, 0, stream>>>(agg, Wrel1, x, Wroot1,
                                                       b1, h, NNODES);
    // Layer 2
    zero_f32_v4<<<zeroBlocks, 256, 0, stream>>>(agg, aggN4);
    scatter_sum<<<scatBlocks, 256, 0, stream>>>(h, src, dst, agg);
    gcn_gemm_wmma<OUTC2, false><<<512, 256, 0, stream>>>(agg, Wrel2, h, Wroot2,
                                                         b2, out, NNODES);
}